// Normalize_77214922048067
// MI455X (gfx1250) — compile-verified
//
#include <hip/hip_runtime.h>
#include <stdint.h>

// Per-row L2 normalize: x [N=1048576, C=128] fp32 -> out same shape.
// Memory bound (1 GB traffic @ 23.3 TB/s ~= 45us). Strategy: stream rows
// through LDS with gfx1250 async global->LDS loads (ASYNCcnt-tracked,
// double buffered), wave-per-row butterfly reduction, one reciprocal per
// row (keeps VALU well under the HBM roofline), NT stores.

typedef float v4f __attribute__((ext_vector_type(4)));
typedef int   v4i __attribute__((ext_vector_type(4)));

#define COLS4 32              // 128 floats = 32 float4 chunks per row
#define WAVES_PER_BLOCK 8     // 256 threads, wave32
#define ROWS_PER_BATCH 4      // rows staged per wave per async batch
#define LDS_V4_PER_WAVE (2 * ROWS_PER_BATCH * COLS4)  // double buffer, 4 KB/wave

#if __has_builtin(__builtin_amdgcn_global_load_async_to_lds_b128)
#define ASYNC_BUILTIN 1
#endif

typedef __attribute__((address_space(1))) v4i* global_v4i_p;
typedef __attribute__((address_space(3))) v4i* lds_v4i_p;

__device__ __forceinline__ void async_load_b128(const v4f* g, v4f* l) {
#ifdef ASYNC_BUILTIN
  __builtin_amdgcn_global_load_async_to_lds_b128(
      (global_v4i_p)g,   // C-style cast: const removal + pointee + addrspace
      (lds_v4i_p)l,
      /*imm offset*/ 0, /*cpol*/ 0);
#else
  unsigned lds_off = (unsigned)(uintptr_t)(lds_v4i_p)l;
  asm volatile("global_load_async_to_lds_b128 %0, %1, off"
               :: "v"(lds_off), "v"(g) : "memory");
#endif
}

// s_wait_asynccnt <= n ; asm so it is also a compiler memory barrier
// (prevents hoisting the LDS reads above the counter wait).
#define WAIT_ASYNC(n) asm volatile("s_wait_asynccnt %0" :: "i"(n) : "memory")

extern "C" __global__ void __launch_bounds__(256)
l2norm_rows_kernel(const float* __restrict__ x, float* __restrict__ out, int nrows) {
  __shared__ v4f smem[WAVES_PER_BLOCK * LDS_V4_PER_WAVE];  // 32 KB

  const int lane  = threadIdx.x & 31;
  const int wave  = threadIdx.x >> 5;
  const int gwave = blockIdx.x * WAVES_PER_BLOCK + wave;
  const int stride = gridDim.x * WAVES_PER_BLOCK * ROWS_PER_BATCH;

  v4f* buf0 = &smem[wave * LDS_V4_PER_WAVE];
  v4f* buf1 = buf0 + ROWS_PER_BATCH * COLS4;

  const v4f* x4 = (const v4f*)x;
  v4f*       o4 = (v4f*)out;

  const int r0 = gwave * ROWS_PER_BATCH;
  if (r0 >= nrows) return;  // wave-uniform; no barriers in this kernel

  // Prime buffer 0 (always issue exactly ROWS_PER_BATCH loads: clamp tail
  // rows so the ASYNCcnt accounting stays exact).
#pragma unroll
  for (int j = 0; j < ROWS_PER_BATCH; ++j) {
    int rr = r0 + j; if (rr >= nrows) rr = nrows - 1;
    async_load_b128(x4 + (size_t)rr * COLS4 + lane, buf0 + j * COLS4 + lane);
  }

  int cur = 0;
  for (int r = r0; r < nrows; r += stride) {
    const int rn = r + stride;
    if (rn < nrows) {
      v4f* nb = cur ? buf0 : buf1;
#pragma unroll
      for (int j = 0; j < ROWS_PER_BATCH; ++j) {
        int rr = rn + j; if (rr >= nrows) rr = nrows - 1;
        async_load_b128(x4 + (size_t)rr * COLS4 + lane, nb + j * COLS4 + lane);
      }
      // 8 async loads outstanding; oldest 4 (current batch) must land.
      WAIT_ASYNC(ROWS_PER_BATCH);
    } else {
      WAIT_ASYNC(0);
    }

    v4f* b = cur ? buf1 : buf0;
#pragma unroll
    for (int j = 0; j < ROWS_PER_BATCH; ++j) {
      const int rr = r + j;
      if (rr >= nrows) break;  // uniform per wave
      v4f v = b[j * COLS4 + lane];                      // ds_load_b128
      float s = v.x * v.x + v.y * v.y + v.z * v.z + v.w * v.w;
      // wave32 butterfly reduction (all lanes end with the row sum)
      s += __shfl_xor(s, 16);
      s += __shfl_xor(s, 8);
      s += __shfl_xor(s, 4);
      s += __shfl_xor(s, 2);
      s += __shfl_xor(s, 1);
      // One divide per row instead of four per lane: keeps aggregate VALU
      // (~25 wave-insts/row) well under the 45us HBM-bound time.
      // s == 0 iff the whole row is zero -> inv = 0 reproduces the
      // reference's where(norm==0, 0, x/norm) exactly (and avoids 0*inf).
      const float inv = (s > 0.0f) ? (1.0f / __builtin_sqrtf(s)) : 0.0f;
      v4f o;
      o.x = v.x * inv; o.y = v.y * inv; o.z = v.z * inv; o.w = v.w * inv;
      // Output (512 MB) exceeds the 192 MB L2; NT store avoids polluting it.
      __builtin_nontemporal_store(o, o4 + (size_t)rr * COLS4 + lane);
    }
    cur ^= 1;
  }
}

extern "C" void kernel_launch(void* const* d_in, const int* in_sizes, int n_in,
                              void* d_out, int out_size, void* d_ws, size_t ws_size,
                              hipStream_t stream) {
  const float* x = (const float*)d_in[0];
  float* out = (float*)d_out;
  const int nrows = in_sizes[0] / 128;
  if (nrows <= 0) return;

  // Each block covers 8 waves * 4 rows = 32 rows per stride step.
  int blocks = (nrows + 31) / 32;
  if (blocks > 4096) blocks = 4096;
  if (blocks < 1) blocks = 1;

  l2norm_rows_kernel<<<blocks, 256, 0, stream>>>(x, out, nrows);
}